// TCN_Parameterized_42305427865833
// MI455X (gfx1250) — compile-verified
//
#include <hip/hip_runtime.h>
#include <stdint.h>

// CDNA5 / gfx1250, wave32. FP32 WMMA: D(16x16) = A(16x4) x B(4x16) + C.
typedef float v2f __attribute__((ext_vector_type(2)));
typedef float v8f __attribute__((ext_vector_type(8)));

#define T_LEN 1024
#define C_CH  128
#define N_SMP 16
#define D_R   320
#define GENLEN 98304

// LDS tile layout: channel pairs interleaved so a v2f B-fragment
// {in[c][t], in[c+1][t]} is one contiguous 8B ds_load_b64.
// slot(c, j) = (c>>1)*PAIR_STRIDE + 2*j + (c&1), j = t - (t0-32), 0..63.
// PAIR_STRIDE = 160 dwords: +160 == +32 (mod 64 banks), so the two lane
// halves (adjacent c-pairs) land on disjoint bank windows -> conflict-free.
#define PAIR_STRIDE 160

__device__ __forceinline__ v8f wmma_f32(v2f a, v2f b, v8f c) {
  // (neg_a, A, neg_b, B, c_mod, C, reuse_a, reuse_b)
  return __builtin_amdgcn_wmma_f32_16x16x4_f32(false, a, false, b, (short)0, c,
                                               false, false);
}

// -------------------------------------------------------------------------
// h[n,c,t] = x[n,t,c]
__global__ __launch_bounds__(256) void transpose_x(float* __restrict__ h,
                                                   const float* __restrict__ x) {
  long idx = (long)blockIdx.x * 256 + threadIdx.x;   // over N*C*T
  long n = idx / (C_CH * T_LEN);
  long rem = idx - n * (C_CH * T_LEN);
  long c = rem / T_LEN;
  long t = rem - c * T_LEN;
  h[idx] = x[n * (long)(T_LEN * C_CH) + t * C_CH + c];
}

// -------------------------------------------------------------------------
// Causal K=2 dilated conv as two GEMMs, WMMA-tiled.
//   out[n,o,t] = act( sum_c W[o,c,0]*in[n,c,t-d] + W[o,c,1]*in[n,c,t] (+b[o]) )
// Epilogue: v = relu(conv + bias); if res: v = relu(v + res).
// Weights laid out (o,c,k): idx = o*256 + c*2 + k; per-sample stride w_sn.
// Block: 256 threads (8 waves); grid (T/32, N). Wave w -> output channels
// [16w,16w+16), 32 timesteps (two 16x16 accumulators). Input tile staged to
// LDS with CDNA5 async global->LDS (ASYNCcnt), shared by all 8 waves.
__global__ __launch_bounds__(256) void conv_k2_wmma(
    float* __restrict__ out,
    const float* __restrict__ in, long in_sn, int in_sc, int in_st,
    const float* __restrict__ w, long w_sn,
    const float* __restrict__ bias,
    const float* __restrict__ res,
    int d)
{
  __shared__ float sIn[(C_CH / 2) * PAIR_STRIDE];   // 64 c-pairs x 160 dw = 40KB
  const int n  = blockIdx.y;
  const int t0 = blockIdx.x * 32;
  const int tid = threadIdx.x;

  // Stage [c][t0-32 .. t0+31] (covers causal tap for d <= 32).
  // Mapping keeps consecutive threads on consecutive LDS dwords.
  const float* inN = in + (long)n * in_sn;
  for (int idx = tid; idx < C_CH * 64; idx += 256) {
    const int p  = idx & 1;          // channel parity
    const int j  = (idx >> 1) & 63;  // time slot within tile
    const int cp = idx >> 7;         // channel pair
    const int c  = 2 * cp + p;
    const int t  = t0 - 32 + j;
    const int slot = cp * PAIR_STRIDE + 2 * j + p;
    if (t >= 0) {
      const float* gptr = inN + (long)c * in_sc + (long)t * in_st;
      const unsigned lds_b = (unsigned)(uintptr_t)&sIn[slot];  // LDS byte addr
      // CDNA5 async DMA: per-lane global -> LDS, tracked by ASYNCcnt.
      asm volatile("global_load_async_to_lds_b32 %0, %1, off"
                   :: "v"(lds_b), "v"(gptr) : "memory");
    } else {
      sIn[slot] = 0.0f;              // causal zero padding
    }
  }
  asm volatile("s_wait_asynccnt 0x0" ::: "memory");
  __syncthreads();

  const int lane  = tid & 31;
  const int laneT = lane & 15;
  const int hi    = lane >> 4;
  const int wave  = tid >> 5;
  const int o0    = wave * 16;

  const float* wN = w + (long)n * w_sn + (long)o0 * 256;

  v8f acc0 = {}, acc1 = {};
  for (int tap = 0; tap < 2; ++tap) {
    const int tshift = (tap == 0) ? -d : 0;
    const int jb0 = (32 + tshift + laneT) * 2;   // dword offset, subtile t0..t0+15
    const int jb1 = jb0 + 32;                    // subtile t0+16..t0+31
#pragma unroll 4
    for (int c0 = 0; c0 < C_CH; c0 += 4) {
      // A (16x4 of W[:, :, tap]): rows M = o0+laneT; K split regs/lane-halves
      const float* wp = wN + (long)laneT * 256 + (c0 + 2 * hi) * 2 + tap;
      v2f a; a.x = wp[0]; a.y = wp[2];
      // B (4x16 of input): one ds_load_b64 per fragment (K pair contiguous)
      const float* rowp = sIn + ((c0 >> 1) + hi) * PAIR_STRIDE;
      v2f b0 = *(const v2f*)(rowp + jb0);
      v2f b1 = *(const v2f*)(rowp + jb1);
      acc0 = wmma_f32(a, b0, acc0);
      acc1 = wmma_f32(a, b1, acc1);
    }
  }

  float* outN = out + (long)n * (C_CH * T_LEN);
  const float* resN = res ? res + (long)n * (C_CH * T_LEN) : nullptr;
#pragma unroll
  for (int rr = 0; rr < 8; ++rr) {
    const int o  = o0 + rr + 8 * hi;
    const int tA = t0 + laneT;
    const int tB = t0 + 16 + laneT;
    const float bv = bias ? bias[o] : 0.0f;
    float vA = fmaxf(acc0[rr] + bv, 0.0f);
    float vB = fmaxf(acc1[rr] + bv, 0.0f);
    if (resN) {
      vA = fmaxf(vA + resN[(long)o * T_LEN + tA], 0.0f);
      vB = fmaxf(vB + resN[(long)o * T_LEN + tB], 0.0f);
    }
    outN[(long)o * T_LEN + tA] = vA;
    outN[(long)o * T_LEN + tB] = vB;
  }
}

// -------------------------------------------------------------------------
// gen[n,g] = r[n,:] . fparam_W[g,:] + fparam_b[g] + last_bias[g]
// M=16 samples (one WMMA tile), K=320, N=98304. Wave handles 32 g.
__global__ __launch_bounds__(256) void gen_gemm_wmma(
    float* __restrict__ gen,
    const float* __restrict__ r,
    const float* __restrict__ fW,
    const float* __restrict__ fb,
    const float* __restrict__ lb)
{
  const int tid   = threadIdx.x;
  const int lane  = tid & 31;
  const int laneT = lane & 15;
  const int hi    = lane >> 4;
  const int wave  = tid >> 5;
  const long g0   = (long)blockIdx.x * 256 + wave * 32;

  const float* rRow = r  + (long)laneT * D_R;            // A: M=n=laneT
  const float* fw0  = fW + (g0 + laneT) * D_R;           // B: N=g columns
  const float* fw1  = fW + (g0 + 16 + laneT) * D_R;

  v8f acc0 = {}, acc1 = {};
#pragma unroll 4
  for (int c0 = 0; c0 < D_R; c0 += 4) {
    const int cb = c0 + 2 * hi;
    v2f a;  a.x  = rRow[cb]; a.y  = rRow[cb + 1];
    v2f b0; b0.x = fw0[cb];  b0.y = fw0[cb + 1];
    v2f b1; b1.x = fw1[cb];  b1.y = fw1[cb + 1];
    acc0 = wmma_f32(a, b0, acc0);
    acc1 = wmma_f32(a, b1, acc1);
  }
#pragma unroll
  for (int rr = 0; rr < 8; ++rr) {
    const int  nn = rr + 8 * hi;
    const long gA = g0 + laneT, gB = g0 + 16 + laneT;
    gen[(long)nn * GENLEN + gA] = acc0[rr] + fb[gA] + lb[gA];
    gen[(long)nn * GENLEN + gB] = acc1[rr] + fb[gB] + lb[gB];
  }
}

// -------------------------------------------------------------------------
// out[n,t,o] = sum_c h[n,c,t] * lin_W[o,c] + lin_b[o]
// A = h tile (M=t, K=c), B = lin_W (K=c, N=o) -> D lands in (t,o) order.
__global__ __launch_bounds__(256) void final_linear_wmma(
    float* __restrict__ out,
    const float* __restrict__ h,
    const float* __restrict__ linW,
    const float* __restrict__ linb)
{
  const int n  = blockIdx.y;
  const int t0 = blockIdx.x * 32;
  const int tid   = threadIdx.x;
  const int lane  = tid & 31;
  const int laneT = lane & 15;
  const int hi    = lane >> 4;
  const int wave  = tid >> 5;
  const int o0    = wave * 16;

  const float* hN = h + (long)n * (C_CH * T_LEN);
  v8f acc0 = {}, acc1 = {};
#pragma unroll 4
  for (int c0 = 0; c0 < C_CH; c0 += 4) {
    const int cb = c0 + 2 * hi;
    v2f a0; a0.x = hN[(long)cb * T_LEN + t0 + laneT];
            a0.y = hN[(long)(cb + 1) * T_LEN + t0 + laneT];
    v2f a1; a1.x = hN[(long)cb * T_LEN + t0 + 16 + laneT];
            a1.y = hN[(long)(cb + 1) * T_LEN + t0 + 16 + laneT];
    const float* wp = linW + (long)(o0 + laneT) * C_CH + cb;
    v2f b;  b.x = wp[0]; b.y = wp[1];
    acc0 = wmma_f32(a0, b, acc0);
    acc1 = wmma_f32(a1, b, acc1);
  }
  float* outN = out + (long)n * (T_LEN * C_CH);
  const int o = o0 + laneT;
  const float bv = linb[o];
#pragma unroll
  for (int rr = 0; rr < 8; ++rr) {
    const int tA = t0 + rr + 8 * hi;
    const int tB = t0 + 16 + rr + 8 * hi;
    outN[(long)tA * C_CH + o] = acc0[rr] + bv;   // 16 lanes -> contiguous o
    outN[(long)tB * C_CH + o] = acc1[rr] + bv;
  }
}

// -------------------------------------------------------------------------
extern "C" void kernel_launch(void* const* d_in, const int* in_sizes, int n_in,
                              void* d_out, int out_size, void* d_ws, size_t ws_size,
                              hipStream_t stream) {
  const float* x         = (const float*)d_in[0];
  const float* r         = (const float*)d_in[1];
  // d_in[2] = f_out_same_in_size (always 1 in setup)
  const float* shared_w1 = (const float*)d_in[3];
  const float* shared_b1 = (const float*)d_in[4];
  const float* shared_w2 = (const float*)d_in[5];
  const float* shared_b2 = (const float*)d_in[6];
  const float* fparam_W  = (const float*)d_in[7];
  const float* fparam_b  = (const float*)d_in[8];
  const float* last_bias = (const float*)d_in[9];
  const float* lin_W     = (const float*)d_in[10];
  const float* lin_b     = (const float*)d_in[11];
  float* out = (float*)d_out;

  float* ws = (float*)d_ws;
  const long HSZ = (long)N_SMP * C_CH * T_LEN;   // 2,097,152 floats
  float* buf0 = ws;
  float* buf1 = ws + HSZ;
  float* buf2 = ws + 2 * HSZ;
  float* gen  = ws + 3 * HSZ;                    // 1,572,864 floats

  const long h_sn = (long)C_CH * T_LEN;          // per-sample stride of h bufs
  const int  h_sc = T_LEN, h_st = 1;

  transpose_x<<<(N_SMP * C_CH * T_LEN) / 256, 256, 0, stream>>>(buf0, x);

  dim3 cgrid(T_LEN / 32, N_SMP);
  float* h = buf0; float* a = buf1; float* o = buf2;

  // 3 shared blocks, d = 1, 2, 4
  for (int i = 0; i < 3; ++i) {
    const int d = 1 << i;
    // a = relu(conv1(h) + b1)
    conv_k2_wmma<<<cgrid, 256, 0, stream>>>(
        a, h, h_sn, h_sc, h_st,
        shared_w1 + (long)i * 32768, 0, shared_b1 + i * C_CH, nullptr, d);
    // h' = relu(relu(conv2(a) + b2) + h)
    conv_k2_wmma<<<cgrid, 256, 0, stream>>>(
        o, a, h_sn, h_sc, h_st,
        shared_w2 + (long)i * 32768, 0, shared_b2 + i * C_CH, h, d);
    float* tmp = h; h = o; o = tmp;              // a is free again next iter
  }

  // hyper-network GEMM: gen (16 x 98304)
  gen_gemm_wmma<<<GENLEN / 256, 256, 0, stream>>>(gen, r, fparam_W, fparam_b,
                                                  last_bias);

  // 3 per-sample-weight blocks, d = 8, 16, 32
  // allw[i][n] lives at gen + i*524288 + n*32768 with (o,c,k) layout.
  for (int i = 0; i < 3; ++i) {
    const int d = 8 << i;
    conv_k2_wmma<<<cgrid, 256, 0, stream>>>(
        o, h, h_sn, h_sc, h_st,
        gen + (long)i * 524288, 32768, nullptr, h, d);
    float* tmp = h; h = o; o = tmp;
  }

  final_linear_wmma<<<cgrid, 256, 0, stream>>>(out, h, lin_W, lin_b);
}